// HardResetSSMBlock_15418932592972
// MI455X (gfx1250) — compile-verified
//
#include <hip/hip_runtime.h>

// ---------------------------------------------------------------------------
// HardResetSSMBlock for MI455X (gfx1250, wave32, WMMA).
//
//   y = silu(x @ W1 + b1) @ W2 + b2, masked per frame, decay on (t+1)%10==0.
//
// Roofline: 17.2 GFLOP / 268 MB mandatory HBM traffic -> ~11.5us floor at
// 23.3 TB/s. FP32 WMMA (16x16x4) keeps exact fp32 math and lands near that
// floor. This revision processes M=32 rows per wave so every B-fragment
// ds_load feeds TWO wmmas -- halving LDS read bandwidth per FLOP, since
// B-fragment traffic (256B/wave/wmma) would otherwise saturate the WGP's
// 256B/clk LDS at ~1 wmma/clk/WGP.
// ---------------------------------------------------------------------------

typedef float v2f __attribute__((ext_vector_type(2)));
typedef float v8f __attribute__((ext_vector_type(8)));

#define S_LEN        16384
#define D_DIM        128
#define N_ROWPAIR    8192           // (B*S)/32 = 262144/32
#define WAVES_PER_WG 8
#define H_STRIDE     132            // padded row stride (floats) -> bank-conflict free
#define H_SLAB       (32 * H_STRIDE)
#define LDS_FLOATS   (16384 + 16384 + WAVES_PER_WG * H_SLAB)   // 66560 (~260 KB)

__global__ __launch_bounds__(256, 1)
void hard_reset_ssm_kernel(const float* __restrict__ x,
                           const unsigned char* __restrict__ mask,
                           const float* __restrict__ W1,
                           const float* __restrict__ b1,
                           const float* __restrict__ W2,
                           const float* __restrict__ b2,
                           float* __restrict__ y)
{
    extern __shared__ float smem[];
    float* sW1 = smem;              // 16384 floats, B-fragment-swizzled W1
    float* sW2 = smem + 16384;      // 16384 floats, B-fragment-swizzled W2
    float* sH  = smem + 32768;      // per-wave 32x132 h slabs

    const int tid  = threadIdx.x;
    const int lane = tid & 31;
    const int wave = tid >> 5;
    const int llo  = lane & 15;     // N / M position within fragment
    const int lhi  = lane >> 4;     // half-wave select

    // ---- One-time: swizzle W1/W2 into exact B-fragment order in LDS. ----
    // Fragment element (kk, ntile, lane, v) holds W[k][n] with
    //   k = kk*4 + 2*(lane>>4) + v,  n = ntile*16 + (lane&15)
    // stored at offset (kk*8 + ntile)*64 + lane*2 + v. A B-operand read is
    // then a single ds_load_b64 per lane hitting all 64 banks.
    for (int o = tid; o < 16384; o += 256) {
        const int v   = o & 1;
        const int ln  = (o >> 1) & 31;
        const int blk = o >> 6;
        const int nt  = blk & 7;
        const int kk  = blk >> 3;
        const int k   = (kk << 2) + ((ln >> 4) << 1) + v;
        const int n   = (nt << 4) + (ln & 15);
        sW1[o] = W1[k * D_DIM + n];
        sW2[o] = W2[k * D_DIM + n];
    }
    __syncthreads();

    // Biases are per-column (N = llo within each 16-wide tile).
    float b1r[8], b2r[8];
#pragma unroll
    for (int nt = 0; nt < 8; ++nt) {
        b1r[nt] = b1[(nt << 4) + llo];
        b2r[nt] = b2[(nt << 4) + llo];
    }

    float* hW = sH + wave * H_SLAB;   // this wave's private 32-row h slab

    const int totalWaves = (int)(gridDim.x) * WAVES_PER_WG;
    const int gw = (int)blockIdx.x * WAVES_PER_WG + wave;

    for (int rp = gw; rp < N_ROWPAIR; rp += totalWaves) {
        const int r0 = rp << 5;    // first of 32 rows handled by this wave

        // ---------------- GEMM1: h = x @ W1 (two 16-row blocks) -------------
        v8f acc[2][8] = {};
        // fp32 A-fragment: lane l reads x[r0 + (l&15)][k], k = kk*4 + 2*lhi + v,
        // v=0,1 contiguous -> one global_load_b64 per K-step per row block.
        const float* xrow0 = x + (size_t)(r0 + llo) * D_DIM + (lhi << 1);
        const float* xrow1 = xrow0 + 16 * D_DIM;
        for (int kk = 0; kk < 32; ++kk) {
            const v2f a0 = *(const v2f*)(xrow0 + (kk << 2));
            const v2f a1 = *(const v2f*)(xrow1 + (kk << 2));
            const float* wb = sW1 + (kk << 9) + (lane << 1);
            v2f bf[8];
#pragma unroll
            for (int nt = 0; nt < 8; ++nt)       // 4x ds_load_2addr_b64, one wait
                bf[nt] = *(const v2f*)(wb + (nt << 6));
#pragma unroll
            for (int nt = 0; nt < 8; ++nt) {     // each B fragment feeds 2 wmmas
                acc[0][nt] = __builtin_amdgcn_wmma_f32_16x16x4_f32(
                    false, a0, false, bf[nt], (short)0, acc[0][nt], false, false);
                acc[1][nt] = __builtin_amdgcn_wmma_f32_16x16x4_f32(
                    false, a1, false, bf[nt], (short)0, acc[1][nt], false, false);
            }
        }

        // ---------------- bias + SiLU, stash h in LDS ----------------
        // D-fragment element e of lane l is h[M = e + 8*lhi][N = nt*16 + llo].
        // Row-major, stride 132 floats: conflict-free writes AND reads
        // (the +16-row offset is 2112 dwords = 0 mod 64 banks).
#pragma unroll
        for (int blk = 0; blk < 2; ++blk) {
#pragma unroll
            for (int nt = 0; nt < 8; ++nt) {
                const float bias = b1r[nt];
#pragma unroll
                for (int e = 0; e < 8; ++e) {
                    const float z = acc[blk][nt][e] + bias;
                    const float s = z * __builtin_amdgcn_rcpf(1.0f + __expf(-z)); // silu
                    hW[((blk << 4) + e + (lhi << 3)) * H_STRIDE + (nt << 4) + llo] = s;
                }
            }
        }

        // ---------------- GEMM2: y = h @ W2 ----------------
        v8f acc2[2][8] = {};
        const float* hrow0 = hW + llo * H_STRIDE + (lhi << 1);
        const float* hrow1 = hrow0 + 16 * H_STRIDE;
        for (int kk = 0; kk < 32; ++kk) {
            const v2f a0 = *(const v2f*)(hrow0 + (kk << 2));   // ds_load_b64
            const v2f a1 = *(const v2f*)(hrow1 + (kk << 2));
            const float* wb = sW2 + (kk << 9) + (lane << 1);
            v2f bf[8];
#pragma unroll
            for (int nt = 0; nt < 8; ++nt)
                bf[nt] = *(const v2f*)(wb + (nt << 6));
#pragma unroll
            for (int nt = 0; nt < 8; ++nt) {
                acc2[0][nt] = __builtin_amdgcn_wmma_f32_16x16x4_f32(
                    false, a0, false, bf[nt], (short)0, acc2[0][nt], false, false);
                acc2[1][nt] = __builtin_amdgcn_wmma_f32_16x16x4_f32(
                    false, a1, false, bf[nt], (short)0, acc2[1][nt], false, false);
            }
        }

        // ---------------- mask * decay factor (per row) ----------------
        float fac[2][8];
#pragma unroll
        for (int blk = 0; blk < 2; ++blk) {
#pragma unroll
            for (int e = 0; e < 8; ++e) {
                const int row = r0 + (blk << 4) + e + (lhi << 3);
                const int s   = row & (S_LEN - 1);            // S is a power of two
                const float sc = (((s + 1) % 10) == 0) ? 0.1f : 1.0f;
                fac[blk][e] = mask[row] ? sc : 0.0f;
            }
        }

        // ---------------- epilogue: bias2, scale, streaming store -----------
#pragma unroll
        for (int blk = 0; blk < 2; ++blk) {
#pragma unroll
            for (int nt = 0; nt < 8; ++nt) {
                const float bias = b2r[nt];
#pragma unroll
                for (int e = 0; e < 8; ++e) {
                    const float val = (acc2[blk][nt][e] + bias) * fac[blk][e];
                    float* dst = y + (size_t)(r0 + (blk << 4) + e + (lhi << 3)) * D_DIM
                               + (nt << 4) + llo;
                    __builtin_nontemporal_store(val, dst);   // y is write-once
                }
            }
        }
    }
}

extern "C" void kernel_launch(void* const* d_in, const int* in_sizes, int n_in,
                              void* d_out, int out_size, void* d_ws, size_t ws_size,
                              hipStream_t stream) {
    const float*         x_   = (const float*)d_in[0];
    const unsigned char* mask = (const unsigned char*)d_in[1];  // jax bool -> 1 byte
    const float*         W1   = (const float*)d_in[2];
    const float*         b1   = (const float*)d_in[3];
    const float*         W2   = (const float*)d_in[4];
    const float*         b2   = (const float*)d_in[5];
    float*               y    = (float*)d_out;

    const size_t shmem = LDS_FLOATS * sizeof(float);  // ~260 KB (< 320 KB WGP LDS)
    (void)hipFuncSetAttribute((const void*)hard_reset_ssm_kernel,
                              hipFuncAttributeMaxDynamicSharedMemorySize,
                              (int)shmem);

    // 512 WGs x 8 waves = 4096 waves; 8192 row-pairs -> 2 per wave.
    hard_reset_ssm_kernel<<<512, 256, shmem, stream>>>(x_, mask, W1, b1, W2, b2, y);
}